// PatchletsExtractorStrided_7078106104297
// MI455X (gfx1250) — compile-verified
//
#include <hip/hip_runtime.h>

#define KNN 16
constexpr int B = 4, T = 32, N = 2048, DF = 16, SEGS = 4, ST = 8;
constexpr int NHALF = N / 2;
constexpr size_t SZ_ITK = (size_t)B * T * N * KNN;       // 4,194,304
constexpr size_t S_IDX0 = 0;
constexpr size_t S_DIST = SZ_ITK;
constexpr size_t S_IDX2 = 2 * SZ_ITK;
constexpr size_t S_PTS  = 3 * SZ_ITK;                    // *3 floats
constexpr size_t S_FEAT = S_PTS + 3 * SZ_ITK;            // *19 floats
constexpr size_t S_NORM = S_FEAT + 19 * SZ_ITK;          // *3 floats
constexpr size_t S_OUTX = S_NORM + 3 * SZ_ITK;           // B*T*N*3
constexpr size_t S_TRAJ = S_OUTX + (size_t)B * T * N * 3;
constexpr int CHAINS = B * SEGS * 2;                     // 32
constexpr size_t CARRY_SZ = (size_t)CHAINS * N * 4;      // padded float4 rows

typedef float v2f __attribute__((ext_vector_type(2)));
typedef float v8f __attribute__((ext_vector_type(8)));

__device__ __forceinline__ void insert16(float (&dl)[KNN], int (&il)[KNN],
                                         float c, int ci) {
  float cur = c; int curi = ci;
#pragma unroll
  for (int t = 0; t < KNN; ++t) {
    bool sw = cur < dl[t];
    float td = dl[t]; int ti = il[t];
    dl[t] = sw ? cur : td; il[t] = sw ? curi : ti;
    cur   = sw ? td : cur; curi = sw ? ti : curi;
  }
}

// Initialize the scan carry: dir0 -> first frame of segment, dir1 -> last frame.
__global__ void init_carry(const float* __restrict__ P, float* __restrict__ carry) {
  int chain = blockIdx.x;
  int b = chain >> 3, seg = (chain >> 1) & 3, dir = chain & 1;
  int frame = seg * ST + (dir ? ST - 1 : 0);
  const float* src = P + ((size_t)(b * T + frame)) * N * 3;
  for (int i = threadIdx.x; i < N; i += blockDim.x) {
    float* dst = carry + ((size_t)chain * N + i) * 4;
    dst[0] = src[i * 3 + 0];
    dst[1] = src[i * 3 + 1];
    dst[2] = src[i * 3 + 2];
    dst[3] = 0.f;
  }
}

// One scan step for all 32 chains. blockIdx.y = chain, blockIdx.x = row block,
// 128 threads = 4 waves, one 16-query tile per wave.
//
// Transposed WMMA mapping: A (16x4, M = candidate points) = (px,py,pz,|p|^2),
// B (4x16, N = queries) = (-2qx,-2qy,-2qz,1), so
//   D[point][query] = |p|^2 - 2 q.p   (order-equivalent to the true distance).
// Lane l holds, in acc[0..7], 8 candidates for query l%16 -> top-k runs
// directly on the accumulator. 2x-unrolled software pipeline with register
// rotation: two accumulators alternate producer/consumer roles so the
// pipeline needs no acc copies.
__global__ __launch_bounds__(128)
void knn_step(const float* __restrict__ P, const float* __restrict__ cin,
              float* __restrict__ cout, float* __restrict__ out, int step) {
  __shared__ float rawS[N * 3];      // async-staged packed xyz (24 KB)
  __shared__ float4 ptsS[N + 32];    // repacked x,y,z,|p|^2 (+2 tile pad)

  int chain = blockIdx.y;
  int b = chain >> 3, seg = (chain >> 1) & 3, dir = chain & 1;
  int frame = dir ? (seg * ST + (ST - 1) - step) : (seg * ST + step);
  const float* pb = P + ((size_t)(b * T + frame)) * N * 3;

  // Stage the candidate frame with the CDNA5 async global->LDS path
  // (ASYNCcnt-tracked, no VGPR data movement).
  {
    auto* raw3 = (__attribute__((address_space(3))) float*)rawS;
    for (int i = threadIdx.x; i < (N * 3) / 4; i += blockDim.x) {
      unsigned ldsoff = (unsigned)(uintptr_t)(raw3 + i * 4);
      unsigned long long ga = (unsigned long long)(uintptr_t)(pb + i * 4);
      asm volatile("global_load_async_to_lds_b128 %0, %1, off"
                   :: "v"(ldsoff), "v"(ga) : "memory");
    }
    asm volatile("s_wait_asynccnt 0x0" ::: "memory");
  }
  __syncthreads();
  for (int i = threadIdx.x; i < N; i += blockDim.x) {
    float x = rawS[i * 3], y = rawS[i * 3 + 1], z = rawS[i * 3 + 2];
    ptsS[i] = make_float4(x, y, z, x * x + y * y + z * z);
  }
  __syncthreads();

  int lane = threadIdx.x & 31, wave = threadIdx.x >> 5;
  int m = lane & 15, hi = lane >> 4;
  int row = (blockIdx.x * 4 + wave) * 16 + m;

  const float4 q = *(const float4*)(cin + ((size_t)chain * N + row) * 4);
  float qx = q.x, qy = q.y, qz = q.z;
  // B-matrix 4x16 (queries): V0 = K0 row (lanes 0-15) / K1 (lanes 16-31);
  //                          V1 = K2 / K3.  Constant across the whole loop.
  v2f bq;
  bq.x = hi ? -2.f * qy : -2.f * qx;
  bq.y = hi ? 1.f       : -2.f * qz;

  float dl[KNN]; int il[KNN];
#pragma unroll
  for (int t = 0; t < KNN; ++t) { dl[t] = 3.402823466e+38f; il[t] = 0; }

  // A-matrix builder: 16x4 (points): V0 = K0 / K2, V1 = K1 / K3.
  auto wmmaTile = [&](int tile_) -> v8f {
    float4 Pp = ptsS[tile_ * 16 + m];
    v2f ap;
    ap.x = hi ? Pp.z : Pp.x;
    ap.y = hi ? Pp.w : Pp.y;
    return __builtin_amdgcn_wmma_f32_16x16x4_f32(
        false, ap, false, bq, (short)0, (v8f){}, false, false);
  };
  // Scan one accumulator; single guard via min-reduction (common case:
  // nothing beats the running 16th-best).
  auto scanAcc = [&](const v8f& acc, int tile_) {
    float mn = fminf(fminf(fminf(acc[0], acc[1]), fminf(acc[2], acc[3])),
                     fminf(fminf(acc[4], acc[5]), fminf(acc[6], acc[7])));
    if (mn < dl[KNN - 1]) {
      int cbase = tile_ * 16 + 8 * hi;
#pragma unroll
      for (int r = 0; r < 8; ++r) {
        float c = acc[r];
        if (c < dl[KNN - 1]) insert16(dl, il, c, cbase + r);
      }
    }
  };

  // 2-deep pipeline with register rotation (pad rows absorb the overrun).
  v8f acc0 = wmmaTile(0), acc1;
  for (int tile = 0; tile < N / 16; tile += 2) {
    acc1 = wmmaTile(tile + 1);
    scanAcc(acc0, tile);
    acc0 = wmmaTile(tile + 2);
    scanAcc(acc1, tile + 1);
  }

  // Merge the two half-lists for each query (wave32 shuffle, lanes l <- l+16).
#pragma unroll
  for (int t = 0; t < KNN; ++t) {
    float c = __shfl(dl[t], m + 16, 32);
    int ci  = __shfl(il[t], m + 16, 32);
    if (hi == 0 && c < dl[KNN - 1]) insert16(dl, il, c, ci);
  }

  if (hi == 0) {
    float q2 = qx * qx + qy * qy + qz * qz;
    float4 P0 = ptsS[il[0]];                 // nearest-neighbor coords = scan carry
    float* co = cout + ((size_t)chain * N + row) * 4;
    co[0] = P0.x; co[1] = P0.y; co[2] = P0.z; co[3] = 0.f;

    if (dir == 0) {                          // out_x comes from the forward pass
      size_t ox = S_OUTX + ((size_t)(b * T + seg * ST + step) * N + row) * 3;
      out[ox] = P0.x; out[ox + 1] = P0.y; out[ox + 2] = P0.z;
    }
    if (row < NHALF) {                       // permuted-concat keeps half of each dir
      int t_out = dir ? (seg * ST + (ST - 1) - step) : (seg * ST + step);
      int n_out = dir ? (NHALF + row) : row;
      size_t base = ((size_t)(b * T + t_out) * N + n_out) * KNN;
#pragma unroll
      for (int t = 0; t < KNN; ++t) {
        float fi = (float)il[t];
        out[S_IDX0 + base + t] = fi;
        out[S_IDX2 + base + t] = fi;
        out[S_DIST + base + t] = dl[t] + q2;
      }
    }
  }
}

// Bandwidth-bound gather/assembly: one thread per (b,t,n,k) element.
__global__ __launch_bounds__(256)
void gather_assemble(const float* __restrict__ P, const float* __restrict__ F,
                     float* __restrict__ out) {
  size_t gid = (size_t)blockIdx.x * blockDim.x + threadIdx.x;
  if (gid >= SZ_ITK) return;
  int k = (int)(gid & (KNN - 1));
  size_t r = gid >> 4;
  int n = (int)(r % N);
  size_t bt = r / N;
  int t = (int)(bt % T), b = (int)(bt / T);

  int idx  = (int)out[S_IDX0 + gid];
  int idx0 = (int)out[S_IDX0 + (r << 4)];

  const float* pr  = P + ((size_t)(b * T + t) * N + idx)  * 3;
  const float* pr0 = P + ((size_t)(b * T + t) * N + idx0) * 3;
  float px = pr[0],  py = pr[1],  pz = pr[2];
  float ox = pr0[0], oy = pr0[1], oz = pr0[2];
  float nx = px - ox, ny = py - oy, nz = pz - oz;

  size_t pbx = S_PTS + gid * 3;
  out[pbx] = px; out[pbx + 1] = py; out[pbx + 2] = pz;
  size_t nb = S_NORM + gid * 3;
  out[nb] = nx; out[nb + 1] = ny; out[nb + 2] = nz;

  // Feature frame: fwd half uses shifted segment frames; bwd half uses the
  // reference's quirk of shifted global frames 0..6.
  int s = t & (ST - 1);
  int ff = (n < NHALF) ? (t - (s ? 1 : 0)) : ((6 - s) > 0 ? (6 - s) : 0);
  const float* fr = F + ((size_t)(b * T + ff) * N + idx) * DF;
  size_t fb = S_FEAT + gid * 19;
#pragma unroll
  for (int c = 0; c < DF; c += 4) {
    float4 v = *(const float4*)(fr + c);
    out[fb + c] = v.x; out[fb + c + 1] = v.y;
    out[fb + c + 2] = v.z; out[fb + c + 3] = v.w;
  }
  out[fb + 16] = nx; out[fb + 17] = ny; out[fb + 18] = nz;

  if (k == 0) {                              // trajectory relative to segment origin
    int t0 = t & ~(ST - 1);
    int idx00 = (int)out[S_IDX0 + (((size_t)(b * T + t0) * N + n) << 4)];
    const float* pr00 = P + ((size_t)(b * T + t0) * N + idx00) * 3;
    size_t tb = S_TRAJ + ((size_t)(b * T + t) * N + n) * 3;
    out[tb]     = ox - pr00[0];
    out[tb + 1] = oy - pr00[1];
    out[tb + 2] = oz - pr00[2];
  }
}

extern "C" void kernel_launch(void* const* d_in, const int* in_sizes, int n_in,
                              void* d_out, int out_size, void* d_ws, size_t ws_size,
                              hipStream_t stream) {
  (void)in_sizes; (void)n_in; (void)out_size; (void)ws_size;
  const float* P = (const float*)d_in[0];
  const float* F = (const float*)d_in[1];
  float* out = (float*)d_out;
  float* ws = (float*)d_ws;
  float* bufA = ws;
  float* bufB = ws + CARRY_SZ;

  init_carry<<<dim3(CHAINS), 256, 0, stream>>>(P, bufA);
  for (int s = 0; s < ST; ++s) {
    const float* cinp = (s & 1) ? bufB : bufA;
    float* coutp      = (s & 1) ? bufA : bufB;
    knn_step<<<dim3(N / 64, CHAINS), 128, 0, stream>>>(P, cinp, coutp, out, s);
  }
  gather_assemble<<<dim3((unsigned)(SZ_ITK / 256)), 256, 0, stream>>>(P, F, out);
}